// Kernel_86337432584781
// MI455X (gfx1250) — compile-verified
//
#include <hip/hip_runtime.h>
#include <hip/hip_bf16.h>

// Problem constants (from reference): N=65536, I=J=16 (IJ=256), Y=9, W=96, H=128
#define HDIM 128
#define WDIM 96
#define KTOT 864      // Y*W = 9*96, GEMM K dimension
#define IJ   256
#define PTS_PER_BLK 64
#define THREADS 256
#define CHUNK_K 12
#define NCHUNK (KTOT / CHUNK_K)   // 72

// LDS strides (bank behavior + alignment):
#define YS_STRIDE  12   // [64][12]: 0..8 = Ysh, 9 = mask, 10 = radius
#define RAD_STRIDE 98   // even (b64-aligned); 98 mod 64 = 34 -> 16 lanes distinct banks
#define QS_STRIDE  18   // even (b64-aligned); 16 consecutive cols hit distinct banks
#define QBUF_WORDS (IJ * QS_STRIDE)   // one Q chunk buffer: 256 cols x 18

typedef float v2f __attribute__((ext_vector_type(2)));
typedef float v8f __attribute__((ext_vector_type(8)));
typedef int   v2i __attribute__((vector_size(8)));   // matches builtin param type

// ---- f32 WMMA --------------------------------
#if __has_builtin(__builtin_amdgcn_wmma_f32_16x16x4_f32)
#define WMMA4(acc, a, b) \
  (acc) = __builtin_amdgcn_wmma_f32_16x16x4_f32(false, (a), false, (b), (short)0, (acc), false, false)
#else
#define WMMA4(acc, a, b) \
  asm("v_wmma_f32_16x16x4_f32 %0, %1, %2, %0" : "+v"(acc) : "v"(a), "v"(b))
#endif

// ---- gfx1250 async global->LDS staging (ASYNCcnt path), guarded ----------
#if __has_builtin(__builtin_amdgcn_global_load_async_to_lds_b64) && \
    __has_builtin(__builtin_amdgcn_s_wait_asynccnt)
#define USE_ASYNC 1
#else
#define USE_ASYNC 0
#endif

#if USE_ASYNC
#define WAIT_ASYNC_PREV() __builtin_amdgcn_s_wait_asynccnt(6)  // keep newest 6 in flight
#define WAIT_ASYNC_ALL()  __builtin_amdgcn_s_wait_asynccnt(0)
#else
#define WAIT_ASYNC_PREV()
#define WAIT_ASYNC_ALL()
#endif

// Stage one chunk row (12 floats, 48B, 8B-aligned) for this thread's column.
__device__ __forceinline__ void stage_chunk(const float* __restrict__ g, float* l) {
#if USE_ASYNC
  // Builtin signature (from hipcc diagnostic): param0 = v2i* in global AS (non-const).
  __attribute__((address_space(1))) v2i* ga =
      (__attribute__((address_space(1))) v2i*)(float*)g;
  __attribute__((address_space(3))) v2i* la =
      (__attribute__((address_space(3))) v2i*)l;
  __builtin_amdgcn_global_load_async_to_lds_b64(ga, la, 0, 0);
  __builtin_amdgcn_global_load_async_to_lds_b64(ga, la, 8, 0);
  __builtin_amdgcn_global_load_async_to_lds_b64(ga, la, 16, 0);
  __builtin_amdgcn_global_load_async_to_lds_b64(ga, la, 24, 0);
  __builtin_amdgcn_global_load_async_to_lds_b64(ga, la, 32, 0);
  __builtin_amdgcn_global_load_async_to_lds_b64(ga, la, 40, 0);
#else
  v2f t0 = ((const v2f*)g)[0], t1 = ((const v2f*)g)[1], t2 = ((const v2f*)g)[2];
  v2f t3 = ((const v2f*)g)[3], t4 = ((const v2f*)g)[4], t5 = ((const v2f*)g)[5];
  ((v2f*)l)[0] = t0; ((v2f*)l)[1] = t1; ((v2f*)l)[2] = t2;
  ((v2f*)l)[3] = t3; ((v2f*)l)[4] = t4; ((v2f*)l)[5] = t5;
#endif
}

__global__ __launch_bounds__(THREADS)
void sph_mlp_contract_kernel(const float* __restrict__ r,
                             const float* __restrict__ Q,   // [16][16][9][96] == Q2[col][k]
                             const float* __restrict__ W1,  // [1][128]
                             const float* __restrict__ b1,  // [128]
                             const float* __restrict__ W2,  // [128][96]
                             const float* __restrict__ b2,  // [96]
                             const float* __restrict__ K0,  // [16][16]
                             float* __restrict__ out,       // [N][16][16]
                             int npts)
{
  __shared__ float YshS[PTS_PER_BLK * YS_STRIDE];   // 3 KiB
  __shared__ float RadS[PTS_PER_BLK * RAD_STRIDE];  // 24.5 KiB
  __shared__ float QsS[2 * QBUF_WORDS];             // 36 KiB double-buffered (total 63.5 KiB)

  const int t = threadIdx.x;
  const int pbase = blockIdx.x * PTS_PER_BLK;

  // ---------- Phase 0: radius, mask, spherical harmonics (1 thread / point) ----------
  if (t < PTS_PER_BLK) {
    const float* rp = r + (size_t)(pbase + t) * 3;
    float x = rp[0], y = rp[1], z = rp[2];
    float rad = sqrtf(x * x + y * y + z * z);
    float maskf = (rad > 0.0f) ? 1.0f : 0.0f;
    float inv = (rad > 0.0f) ? (1.0f / rad) : 1.0f;   // safe = where(mask, radii, 1)
    float dx = x * inv, dy = y * inv, dz = z * inv;
    const float SQ3  = 1.7320508075688772f;
    const float SQ5  = 2.2360679774997896f;
    const float SQ15 = 3.8729833462074170f;
    float* yr = YshS + t * YS_STRIDE;
    yr[0] = 1.0f;
    yr[1] = SQ3 * dy;
    yr[2] = SQ3 * dz;
    yr[3] = SQ3 * dx;
    yr[4] = SQ15 * dx * dy;
    yr[5] = SQ15 * dy * dz;
    yr[6] = 0.5f * SQ5 * (3.0f * dz * dz - 1.0f);
    yr[7] = SQ15 * dx * dz;
    yr[8] = 0.5f * SQ15 * (dx * dx - dy * dy);
    yr[9] = maskf;
    yr[10] = rad;
    yr[11] = 0.0f;
  }
  __syncthreads();

  // ---------- Phase 1: MLP radial net -> RadS (4 threads / point, 24 outputs each) ----
  {
    const int p  = t >> 2;
    const int wq = (t & 3) * 24;
    const float rad = YshS[p * YS_STRIDE + 10];
    float acc[24];
#pragma unroll
    for (int k = 0; k < 24; ++k) acc[k] = b2[wq + k];
    for (int j = 0; j < HDIM; ++j) {
      float h = __builtin_fmaf(rad, W1[j], b1[j]);
      h = fmaxf(h, 0.0f);
      const float* w2r = W2 + j * WDIM + wq;   // 16B aligned, contiguous 24 -> b128 loads
#pragma unroll
      for (int k = 0; k < 24; ++k) acc[k] = __builtin_fmaf(h, w2r[k], acc[k]);
    }
    float* dst = RadS + p * RAD_STRIDE + wq;
#pragma unroll
    for (int k = 0; k < 24; ++k) dst[k] = acc[k];
  }
  __syncthreads();

  // ---------- Phase 2: C[64 x 256] += P[64 x 864] * Q2^T via V_WMMA_F32_16X16X4_F32 ----
  // 8 waves; wave = (mtile 0..3) x (N-half 0..1); each wave owns 8 N-tiles.
  const int wv   = t >> 5;
  const int lane = t & 31;
  const int mt   = wv & 3;        // M tile (16 points)
  const int nh   = wv >> 2;       // which 8 N-tiles
  const int ln   = lane & 15;
  const int kg   = lane >> 4;     // K half within a K=4 step
  const int mloc = mt * 16 + ln;  // A-matrix row = local point index

  v8f acc[8];
  {
    v8f zero = {};
#pragma unroll
    for (int i = 0; i < 8; ++i) acc[i] = zero;
  }

  const float* gQ = Q + (size_t)t * KTOT;      // this thread's column of Q2
  float* qStage = QsS + t * QS_STRIDE;

  // Prologue: prefetch chunk 0 into buffer 0.
  stage_chunk(gQ, qStage);

  for (int c = 0; c < NCHUNK; ++c) {
    // Prefetch chunk c+1 into the other buffer (its previous readers finished at the
    // end-of-iteration barrier of c-1), overlapping this chunk's WMMA work.
    if (c + 1 < NCHUNK) {
      stage_chunk(gQ + (c + 1) * CHUNK_K, qStage + ((c + 1) & 1) * QBUF_WORDS);
      WAIT_ASYNC_PREV();   // async loads complete in order: oldest 6 (chunk c) done
    } else {
      WAIT_ASYNC_ALL();
    }
    __syncthreads();       // chunk c visible to all waves

    const float* qb = QsS + (c & 1) * QBUF_WORDS;
    const int yidx  = c >> 3;             // 12 | 96 -> y constant per chunk
    const int wbase = (c & 7) * CHUNK_K;  // w offset of this chunk within its y block
    const float ym = YshS[mloc * YS_STRIDE + yidx];
    const float* qcb = qb + (nh * 128 + ln) * QS_STRIDE;  // col base = (nh*8)*16 + ln

#pragma unroll
    for (int s = 0; s < 3; ++s) {
      const int kl = 4 * s + 2 * kg;
      // A fragment on the fly: P[m,k] = Ysh[m,y] * Rad[m,w]
      const v2f rp = *(const v2f*)(RadS + mloc * RAD_STRIDE + wbase + kl);
      v2f a;
      a.x = ym * rp.x;
      a.y = ym * rp.y;
      // Batch all 8 B fragments first so DS loads clause up with a single wait.
      v2f bf[8];
#pragma unroll
      for (int n8 = 0; n8 < 8; ++n8)
        bf[n8] = *(const v2f*)(qcb + n8 * 16 * QS_STRIDE + kl);
#pragma unroll
      for (int n8 = 0; n8 < 8; ++n8)
        WMMA4(acc[n8], a, bf[n8]);
    }
    __syncthreads();       // done reading buffer (c&1) before it's refilled next iter
  }

  // ---------- Epilogue: out = mask ? k1 : K0 (C/D layout: VGPR rr -> M = rr + 8*kg) ----
  float k0v[8];
#pragma unroll
  for (int n8 = 0; n8 < 8; ++n8) k0v[n8] = K0[(nh * 8 + n8) * 16 + ln];
  float msk[8];
#pragma unroll
  for (int rr = 0; rr < 8; ++rr) msk[rr] = YshS[(mt * 16 + rr + 8 * kg) * YS_STRIDE + 9];

#pragma unroll
  for (int n8 = 0; n8 < 8; ++n8) {
    const int col = (nh * 8 + n8) * 16 + ln;
#pragma unroll
    for (int rr = 0; rr < 8; ++rr) {
      const int plocal = mt * 16 + rr + 8 * kg;
      const float v = acc[n8][rr];
      out[(size_t)(pbase + plocal) * IJ + col] = (msk[rr] > 0.0f) ? v : k0v[n8];
    }
  }
}

extern "C" void kernel_launch(void* const* d_in, const int* in_sizes, int n_in,
                              void* d_out, int out_size, void* d_ws, size_t ws_size,
                              hipStream_t stream) {
  const float* r  = (const float*)d_in[0];
  const float* Q  = (const float*)d_in[1];
  const float* W1 = (const float*)d_in[2];
  const float* b1 = (const float*)d_in[3];
  const float* W2 = (const float*)d_in[4];
  const float* b2 = (const float*)d_in[5];
  const float* K0 = (const float*)d_in[6];
  float* out = (float*)d_out;

  const int npts = in_sizes[0] / 3;              // 65536
  dim3 grid(npts / PTS_PER_BLK);                 // 1024 WGs
  sph_mlp_contract_kernel<<<grid, THREADS, 0, stream>>>(r, Q, W1, b1, W2, b2, K0, out, npts);
}